// DocumentDualEmbedder_55439437856849
// MI455X (gfx1250) — compile-verified
//
#include <hip/hip_runtime.h>

// Problem constants (match reference).
#define B_  64
#define S_  2048
#define D_  256
#define VOCAB_ 32000
#define NCH 16                   // s-chunks per document
#define SCHUNK (S_ / NCH)        // 128 rows per chunk

typedef __attribute__((ext_vector_type(2))) float v2f;
typedef __attribute__((ext_vector_type(8))) float v8f;

// ---------------------------------------------------------------------------
// Phase 1: w[b,s] = idf[chunk[b,s]]
// ---------------------------------------------------------------------------
__global__ void k_weights(const int* __restrict__ chunk,
                          const float* __restrict__ idf,
                          float* __restrict__ w, int n) {
  int i = blockIdx.x * blockDim.x + threadIdx.x;
  if (i < n) w[i] = idf[chunk[i]];
}

// ---------------------------------------------------------------------------
// Phase 1b: wtot[b] = sum_s w[b,s]  (deterministic LDS tree reduction)
// ---------------------------------------------------------------------------
__global__ void k_wtot(const float* __restrict__ w, float* __restrict__ wtot) {
  __shared__ float red[256];
  const int b = blockIdx.x, t = threadIdx.x;
  float s = 0.f;
  for (int i = t; i < S_; i += 256) s += w[(size_t)b * S_ + i];
  red[t] = s;
  __syncthreads();
  for (int off = 128; off > 0; off >>= 1) {
    if (t < off) red[t] += red[t + off];
    __syncthreads();
  }
  if (t == 0) wtot[b] = red[0];
}

// ---------------------------------------------------------------------------
// Phase 2: fused streaming pass over encoding.
// Grid: (NCH, B). Block: 512 threads = 16 waves; wave i owns columns
// d = 16*i .. 16*i+15.  Per K=4 step each wave issues two
// V_WMMA_F32_16X16X4_F32:
//   A row0 = w-slice, A row1 = ones, rows 2..15 = 0
//   WMMA #1: B = E tile      -> C row0 = sum(w*E), C row1 = sum(E)
//   WMMA #2: B = E*E tile    -> C row1 = sum(E^2)
// Max/min tracked in VALU; halves combined with shfl_xor(16).
//
// 32-bit A(16x4)/B(4x16) VGPR layout (ISA 7.12.2): VGPR0 holds K=0 for
// lanes 0-15 and K=2 for lanes 16-31; VGPR1 holds K=1 / K=3.  So lane L
// (hl = L>>4, n = L&15) loads E rows k+2*hl and k+2*hl+1, column dbase+n.
//
// A-matrix values are built branchlessly: every lane loads the (broadcast)
// w pair for its two rows, then v_cndmask selects {w, 1, 0} by lane row.
// ---------------------------------------------------------------------------
__global__ void __launch_bounds__(512)
k_main(const float* __restrict__ enc, const float* __restrict__ w,
       float* __restrict__ pWsum, float* __restrict__ pSum,
       float* __restrict__ pSq,   float* __restrict__ pMax,
       float* __restrict__ pMin) {
  const int b    = blockIdx.y;
  const int ch   = blockIdx.x;
  const int s0   = ch * SCHUNK;
  const int wave = threadIdx.x >> 5;
  const int lane = threadIdx.x & 31;
  const int n    = lane & 15;
  const int hl   = lane >> 4;
  const int dbase = wave << 4;

  const float* __restrict__ Eb = enc + ((size_t)b * S_ + s0) * D_;
  const float* __restrict__ wb = w + (size_t)b * S_ + s0;

  v8f c  = {};   // row0: sum(w*E), row1: sum(E)
  v8f c2 = {};   // row1: sum(E^2)
  float vmax = -__builtin_inff();
  float vmin =  __builtin_inff();

  const bool  isw  = (n == 0);            // lane belongs to A row 0 (weights)
  const float ones = (n == 1) ? 1.f : 0.f; // A row 1 = ones, rows 2..15 = 0

#pragma unroll 4
  for (int k = 0; k < SCHUNK; k += 4) {
    const int r0 = k + 2 * hl;           // always even
    const float b0 = Eb[(size_t)r0 * D_ + dbase + n];
    const float b1 = Eb[(size_t)(r0 + 1) * D_ + dbase + n];
    const v2f wp = *(const v2f*)(wb + r0);  // broadcast within half-wave

    vmax = fmaxf(vmax, fmaxf(b0, b1));
    vmin = fminf(vmin, fminf(b0, b1));

    const float a0 = isw ? wp[0] : ones;    // v_cndmask, no branches
    const float a1 = isw ? wp[1] : ones;

    v2f A  = {a0, a1};
    v2f Bv = {b0, b1};
    c = __builtin_amdgcn_wmma_f32_16x16x4_f32(false, A, false, Bv,
                                              (short)0, c, false, false);
    v2f B2 = {b0 * b0, b1 * b1};
    c2 = __builtin_amdgcn_wmma_f32_16x16x4_f32(false, A, false, B2,
                                               (short)0, c2, false, false);
  }

  // Each half-wave saw rows {0,1} vs {2,3} mod 4 of the chunk: combine.
  const float omax = fmaxf(vmax, __shfl_xor(vmax, 16, 32));
  const float omin = fminf(vmin, __shfl_xor(vmin, 16, 32));

  if (lane < 16) {  // C-matrix: VGPR r, lanes 0-15 hold M=r, N=lane
    const size_t idx = (((size_t)b * NCH + ch) * D_) + dbase + n;
    pWsum[idx] = c[0];
    pSum[idx]  = c[1];
    pSq[idx]   = c2[1];
    pMax[idx]  = omax;
    pMin[idx]  = omin;
  }
}

// ---------------------------------------------------------------------------
// Phase 3: reduce chunk partials, finalize [B, 4*D] output.
// ---------------------------------------------------------------------------
__global__ void k_finalize(const float* __restrict__ pWsum,
                           const float* __restrict__ pSum,
                           const float* __restrict__ pSq,
                           const float* __restrict__ pMax,
                           const float* __restrict__ pMin,
                           const float* __restrict__ wtot,
                           float* __restrict__ out) {
  const int b = blockIdx.x;
  const int d = threadIdx.x;  // 256 threads == D_
  float ws = 0.f, sm = 0.f, sq = 0.f;
  float mx = -__builtin_inff(), mn = __builtin_inff();
  for (int ch = 0; ch < NCH; ++ch) {
    const size_t idx = (((size_t)b * NCH + ch) * D_) + d;
    ws += pWsum[idx];
    sm += pSum[idx];
    sq += pSq[idx];
    mx = fmaxf(mx, pMax[idx]);
    mn = fminf(mn, pMin[idx]);
  }
  const float mean = ws / wtot[b];
  float var = (sq - sm * sm / (float)S_) / (float)(S_ - 1);
  var = fmaxf(var, 0.f);
  float* ob = out + (size_t)b * (4 * D_);
  ob[d]            = mean;
  ob[D_ + d]       = mx;
  ob[2 * D_ + d]   = mn;
  ob[3 * D_ + d]   = __builtin_sqrtf(var);
}

// ---------------------------------------------------------------------------
extern "C" void kernel_launch(void* const* d_in, const int* in_sizes, int n_in,
                              void* d_out, int out_size, void* d_ws, size_t ws_size,
                              hipStream_t stream) {
  (void)in_sizes; (void)n_in; (void)out_size; (void)ws_size;
  const int*   chunk = (const int*)d_in[0];     // [B, S] int32
  const float* enc   = (const float*)d_in[1];   // [B, S, D] f32
  const float* idf   = (const float*)d_in[2];   // [VOCAB] f32
  float* out = (float*)d_out;                   // [B, 4*D] f32
  float* ws  = (float*)d_ws;

  // Workspace layout (floats): w[B*S] | wtot[64] | 5 x partial[B*NCH*D]
  float* w    = ws;
  float* wtot = ws + (size_t)B_ * S_;
  float* p    = wtot + 64;
  const size_t psz = (size_t)B_ * NCH * D_;
  float* pWsum = p;
  float* pSum  = p + psz;
  float* pSq   = p + 2 * psz;
  float* pMax  = p + 3 * psz;
  float* pMin  = p + 4 * psz;

  const int n = B_ * S_;
  k_weights<<<(n + 255) / 256, 256, 0, stream>>>(chunk, idf, w, n);
  k_wtot<<<B_, 256, 0, stream>>>(w, wtot);
  k_main<<<dim3(NCH, B_), 512, 0, stream>>>(enc, w, pWsum, pSum, pSq, pMax, pMin);
  k_finalize<<<B_, 256, 0, stream>>>(pWsum, pSum, pSq, pMax, pMin, wtot, out);
}